// MLP_55035710931773
// MI455X (gfx1250) — compile-verified
//
#include <hip/hip_runtime.h>

// Greedy codebook ordering (N=4096) for MI455X / gfx1250.
//
//  * 6.9e10 fp32 FLOPs; D (64MB) L2-resident (192MB L2) -> WMMA compute-bound.
//  * 256 blocks of 16 steps: old-column contributions = dense N x 16 GEMM with
//    K = t0 on V_WMMA_F32_16X16X4_F32 (fp32-exact).
//  * Branch-free inner loop: zero-row-padded D^T kills the j<t0 guard; a
//    REVERSED zero-guarded weight table in LDS makes every B pair an ascending
//    ds_load_2addr_b32 that lands directly in the operand register pair.
//  * 4 independent WMMA chains per wave; 128 blocks x (2 tiles x 4-way K
//    split) with LDS reduction for WGP occupancy.
//  * In-block sequential coupling (rank-1 update + global argmin) in a single
//    1024-thread workgroup kernel; kernel-boundary sync (graph-capture safe).

#define N_CB 4096
#define ZROW N_CB  // index of the appended all-zero row of DT

typedef __attribute__((ext_vector_type(2))) float v2f;
typedef __attribute__((ext_vector_type(8))) float v8f;

// ---------------------------------------------------------------------------
// Init: idx[0]=start, selection mask, weight passthrough, zero pad-row of DT.
// ---------------------------------------------------------------------------
__global__ void init_kernel(const int* __restrict__ index_ls,
                            const float* __restrict__ weight,
                            int* __restrict__ idx_out,
                            float* __restrict__ w_out,
                            int* __restrict__ sel,
                            float* __restrict__ DT) {
  int tid = blockIdx.x * blockDim.x + threadIdx.x;
  if (tid >= N_CB) return;
  int start = index_ls[0];
  idx_out[tid] = (tid == 0) ? start : 0;
  sel[tid] = (tid == start) ? 1 : 0;
  if (tid < N_CB - 1) w_out[tid] = weight[tid];
  DT[(size_t)ZROW * N_CB + tid] = 0.0f;   // zero row for K-padding
}

// ---------------------------------------------------------------------------
// Transpose D -> DT (row-major rows = columns of D) for coalesced gathers.
// ---------------------------------------------------------------------------
__global__ void transpose_kernel(const float* __restrict__ D,
                                 float* __restrict__ DT) {
  __shared__ float tile[32][33];
  int bx = blockIdx.x * 32, by = blockIdx.y * 32;
  int tx = threadIdx.x, ty = threadIdx.y;
#pragma unroll
  for (int i = 0; i < 32; i += 8)
    tile[ty + i][tx] = D[(by + ty + i) * N_CB + (bx + tx)];
  __syncthreads();
#pragma unroll
  for (int i = 0; i < 32; i += 8)
    DT[(bx + ty + i) * N_CB + (by + tx)] = tile[tx][ty + i];
}

// ---------------------------------------------------------------------------
// Old-column GEMM for the 16-step block starting at t0:
//   corr[m][c] = sum_{j<t0} DT[idx[j]][c] * w[t0+m-1-j],  m = 0..15
// 128 blocks x 8 waves: 2 row-tiles per block, 4-way K split across waves.
// A 16x4 f32 layout: M = lane&15; VGPR0 = K{0|2}, VGPR1 = K{1|3} per half.
// B 4x16: same K split, N = lane&15; coef(j) = swzr[j + 31 - m + 2h].
// ---------------------------------------------------------------------------
__global__ __launch_bounds__(256) void gemm_kernel(
    const float* __restrict__ DT, const int* __restrict__ idxbuf,
    const float* __restrict__ w, float* __restrict__ corr, int t0) {
  __shared__ int sidx[N_CB + 16];
  __shared__ float swzr[N_CB + 64];
  __shared__ float sacc[2][3][256];

  int Kr = (t0 + 15) & ~15;               // K extent, unroll-16 granularity
  for (int j = threadIdx.x; j < Kr; j += 256)
    sidx[j] = (j < t0) ? idxbuf[j] : ZROW;          // pad -> zero row of DT
  int nswz = t0 + 48;                      // swzr[i] = w[t0+30-i], guarded
  for (int i = threadIdx.x; i < nswz; i += 256) {
    int x = t0 + 30 - i;
    swzr[i] = (x >= 0 && x < N_CB - 1) ? w[x] : 0.0f;
  }
  __syncthreads();

  int wid = threadIdx.x >> 5;
  int lane = threadIdx.x & 31;
  int tl = wid & 1;                        // local tile 0..1
  int kq = wid >> 1;                       // K quarter 0..3
  int tile = blockIdx.x * 2 + tl;          // 128 * 2 = 256 row tiles
  int m = lane & 15;
  int h = lane >> 4;
  int row = tile * 16 + m;
  int wb = 31 - m + 2 * h;                 // coef(j) = swzr[j + wb] (ascending)

  int Kq = ((Kr >> 2) + 15) & ~15;
  int k0 = kq * Kq;
  int k1 = min(Kr, k0 + Kq);

  v8f c0 = {}, c1 = {}, c2 = {}, c3 = {};
  for (int k = k0; k < k1; k += 16) {
    int j = k + 2 * h;
    int p = k + wb;
    // issue all loads first, then 4 independent WMMA chains
    v2f a0, a1, a2, a3, b0, b1, b2, b3;
    a0.x = DT[sidx[j] * N_CB + row];
    a0.y = DT[sidx[j + 1] * N_CB + row];
    a1.x = DT[sidx[j + 4] * N_CB + row];
    a1.y = DT[sidx[j + 5] * N_CB + row];
    a2.x = DT[sidx[j + 8] * N_CB + row];
    a2.y = DT[sidx[j + 9] * N_CB + row];
    a3.x = DT[sidx[j + 12] * N_CB + row];
    a3.y = DT[sidx[j + 13] * N_CB + row];
    b0.x = swzr[p];
    b0.y = swzr[p + 1];
    b1.x = swzr[p + 4];
    b1.y = swzr[p + 5];
    b2.x = swzr[p + 8];
    b2.y = swzr[p + 9];
    b3.x = swzr[p + 12];
    b3.y = swzr[p + 13];
    c0 = __builtin_amdgcn_wmma_f32_16x16x4_f32(false, a0, false, b0, (short)0,
                                               c0, false, false);
    c1 = __builtin_amdgcn_wmma_f32_16x16x4_f32(false, a1, false, b1, (short)0,
                                               c1, false, false);
    c2 = __builtin_amdgcn_wmma_f32_16x16x4_f32(false, a2, false, b2, (short)0,
                                               c2, false, false);
    c3 = __builtin_amdgcn_wmma_f32_16x16x4_f32(false, a3, false, b3, (short)0,
                                               c3, false, false);
  }
  v8f acc = (c0 + c1) + (c2 + c3);

  // K-split reduction in LDS, then store corr[step][row] (step-major).
  if (kq != 0) {
#pragma unroll
    for (int v = 0; v < 8; ++v) sacc[tl][kq - 1][lane * 8 + v] = acc[v];
  }
  __syncthreads();
  if (kq == 0) {
#pragma unroll
    for (int v = 0; v < 8; ++v)
      acc[v] += (sacc[tl][0][lane * 8 + v] + sacc[tl][1][lane * 8 + v]) +
                sacc[tl][2][lane * 8 + v];
#pragma unroll
    for (int v = 0; v < 8; ++v)
      corr[m * N_CB + tile * 16 + v + 8 * h] = acc[v];
  }
}

// ---------------------------------------------------------------------------
// 16 sequential greedy steps (single 1024-thread workgroup):
// masked scan + first-min argmin (jnp.argmin tie-break) + rank-1 update.
// corr is step-major -> fully coalesced scan and update.
// ---------------------------------------------------------------------------
__global__ __launch_bounds__(1024) void step_kernel(
    float* __restrict__ corr, const float* __restrict__ DT,
    int* __restrict__ idxbuf, int* __restrict__ sel,
    const float* __restrict__ w, const float* __restrict__ bias, int t0) {
  __shared__ float swv[32];
  __shared__ int swc[32];
  __shared__ int sbest;
  const float INF = __builtin_inff();
  int tid = threadIdx.x;
  int lane = tid & 31;
  int wid = tid >> 5;
  float bb = bias[0];
  int nsteps = min(16, N_CB - t0);

  for (int s = 0; s < nsteps; ++s) {
    const float* cs = corr + s * N_CB;
    float bv = INF;
    int bc = 0x7fffffff;
    for (int c = tid; c < N_CB; c += 1024) {
      float v = fmaxf(cs[c] + bb, 0.0f);
      if (v == 0.0f || sel[c]) v = INF;
      if (v < bv || (v == bv && c < bc)) { bv = v; bc = c; }
    }
#pragma unroll
    for (int off = 16; off > 0; off >>= 1) {
      float ov = __shfl_down(bv, off, 32);
      int oc = __shfl_down(bc, off, 32);
      if (ov < bv || (ov == bv && oc < bc)) { bv = ov; bc = oc; }
    }
    if (lane == 0) { swv[wid] = bv; swc[wid] = bc; }
    __syncthreads();
    if (wid == 0) {
      bv = swv[lane];
      bc = swc[lane];
#pragma unroll
      for (int off = 16; off > 0; off >>= 1) {
        float ov = __shfl_down(bv, off, 32);
        int oc = __shfl_down(bc, off, 32);
        if (ov < bv || (ov == bv && oc < bc)) { bv = ov; bc = oc; }
      }
      if (lane == 0) {
        sbest = bc;
        idxbuf[t0 + s] = bc;
        sel[bc] = 1;
      }
    }
    __syncthreads();
    int bi = sbest;
    const float* dtr = DT + bi * N_CB;
    for (int c = tid; c < N_CB; c += 1024) {
      float dv = dtr[c];
      for (int mm = s + 1; mm < nsteps; ++mm)
        corr[mm * N_CB + c] += w[mm - 1 - s] * dv;
    }
    __syncthreads();
  }
}

// ---------------------------------------------------------------------------
extern "C" void kernel_launch(void* const* d_in, const int* in_sizes, int n_in,
                              void* d_out, int out_size, void* d_ws,
                              size_t ws_size, hipStream_t stream) {
  (void)in_sizes; (void)n_in; (void)out_size; (void)ws_size;

  const int* index_ls = (const int*)d_in[0];
  const float* D = (const float*)d_in[1];
  const float* w = (const float*)d_in[2];
  const float* bias = (const float*)d_in[3];

  int* idx_out = (int*)d_out;              // 4096 int32 indices
  float* w_out = (float*)d_out + N_CB;     // 4095 passthrough weights

  // Workspace: DT (4097 rows x 4096, 64MB+16KB) | corr (16x4096) | sel (4096)
  char* ws = (char*)d_ws;
  float* DT = (float*)ws;
  size_t dt_bytes = (size_t)(N_CB + 1) * N_CB * sizeof(float);
  float* corr = (float*)(ws + dt_bytes);
  int* sel = (int*)(ws + dt_bytes + (size_t)16 * N_CB * sizeof(float));

  init_kernel<<<N_CB / 256, 256, 0, stream>>>(index_ls, w, idx_out, w_out, sel,
                                              DT);
  transpose_kernel<<<dim3(N_CB / 32, N_CB / 32), dim3(32, 8), 0, stream>>>(D,
                                                                           DT);
  for (int b = 0; b < N_CB / 16; ++b) {
    int t0 = 1 + 16 * b;                   // columns selected before block
    gemm_kernel<<<128, 256, 0, stream>>>(DT, idx_out, w, corr, t0);
    step_kernel<<<1, 1024, 0, stream>>>(corr, DT, idx_out, sel, w, bias, t0);
  }
}